// MiningGraphNet_51548197487013
// MI455X (gfx1250) — compile-verified
//
#include <hip/hip_runtime.h>

typedef __attribute__((ext_vector_type(2))) float v2f;
typedef __attribute__((ext_vector_type(4))) float v4f;
typedef __attribute__((ext_vector_type(8))) float v8f;

#define WAVES_PER_BLOCK 4
#define MT_PER_WAVE 4

__device__ inline void atomicAddF32(float* p, float v) {
  // relaxed agent-scope fadd -> global_atomic_add_f32 (no CAS loop)
  __hip_atomic_fetch_add(p, v, __ATOMIC_RELAXED, __HIP_MEMORY_SCOPE_AGENT);
}

__global__ void fill_zero_kernel(float* __restrict__ p, long long n) {
  long long i = (long long)blockIdx.x * blockDim.x + threadIdx.x;
  if (i < n) p[i] = 0.0f;
}

__global__ void deg_kernel(const int* __restrict__ src, const float* __restrict__ ea,
                           float* __restrict__ deg, int E) {
  int e = blockIdx.x * blockDim.x + threadIdx.x;
  if (e < E) atomicAddF32(&deg[src[e]], ea[e]);
}

__global__ void dis_kernel(float* __restrict__ deg, int N) {
  int i = blockIdx.x * blockDim.x + threadIdx.x;
  if (i < N) {
    float d = deg[i];
    deg[i] = (d > 0.0f) ? rsqrtf(d) : 0.0f;   // in-place deg -> D^{-1/2}
  }
}

__global__ void norm_kernel(const int* __restrict__ src, const int* __restrict__ dst,
                            const float* __restrict__ ea, const float* __restrict__ dis,
                            float* __restrict__ norm, int E) {
  int e = blockIdx.x * blockDim.x + threadIdx.x;
  if (e < E) norm[e] = -dis[src[e]] * ea[e] * dis[dst[e]];
}

// F == 1 path: one thread per edge.
__global__ void propagate_f1_kernel(const float* __restrict__ x,
                                    const int* __restrict__ src, const int* __restrict__ dst,
                                    const float* __restrict__ norm,
                                    float* __restrict__ out, int E) {
  int e = blockIdx.x * blockDim.x + threadIdx.x;
  if (e >= E) return;
  atomicAddF32(&out[dst[e]], norm[e] * x[src[e]]);
}

// F % 4 == 0 path: one thread per (edge, 4 features). LOG4 = log2(F) - 2.
// Consecutive threads cover one edge's feature row -> coalesced b128 gather.
__global__ void propagate_vec4_kernel(const float* __restrict__ x,
                                      const int* __restrict__ src, const int* __restrict__ dst,
                                      const float* __restrict__ norm,
                                      float* __restrict__ out, int E, int F, int LOG4) {
  long long tid = (long long)blockIdx.x * blockDim.x + threadIdx.x;
  long long total = ((long long)E << LOG4);
  if (tid >= total) return;
  int e  = (int)(tid >> LOG4);
  int f0 = ((int)tid & ((1 << LOG4) - 1)) << 2;
  float w = norm[e];
  const v4f xv = *(const v4f*)(x + (long long)src[e] * F + f0);
  float* o = out + (long long)dst[e] * F + f0;
  atomicAddF32(o + 0, w * xv.x);
  atomicAddF32(o + 1, w * xv.y);
  atomicAddF32(o + 2, w * xv.z);
  atomicAddF32(o + 3, w * xv.w);
}

// t2 = 2*t2 - t0   (Chebyshev recurrence finish)
__global__ void tx2_finish_kernel(float* __restrict__ t2, const float* __restrict__ t0,
                                  long long n) {
  long long i = (long long)blockIdx.x * blockDim.x + threadIdx.x;
  if (i < n) t2[i] = 2.0f * t2[i] - t0[i];
}

// ---------- WMMA GEMM, compile-time specialized ----------
// V_WMMA_F32_16X16X4_F32 fragment layouts (wave32):
//  A 16x4: lanes 0-15 hold K={kb,kb+1}, lanes 16-31 hold K={kb+2,kb+3}; M = lane%16.
//  B 4x16: mirrored, N = lane%16 across both lane halves.
//  C/D 16x16: vgpr r, lanes 0-15 -> M=r, lanes 16-31 -> M=8+r; N = lane%16.

template <int Fin>
__device__ inline v2f load_afrag(const float* __restrict__ X, int arow, int kt, int half) {
  if constexpr (Fin % 4 == 0) {
    const int kb = kt * 4 + 2 * half;
    return *(const v2f*)(X + (long long)arow * Fin + kb);   // one global_load_b64
  } else {  // Fin == 1: pad K=1..3 with zero (branchless select)
    const float v = X[arow];
    v2f a;
    a.x = (half == 0) ? v : 0.0f;
    a.y = 0.0f;
    return a;
  }
}

template <int Fin, int Fout>
__device__ inline v2f load_bfrag(const float* __restrict__ Wt, int l16, int half,
                                 int kt, int col0) {
  v2f b;
  if constexpr (Fin % 4 == 0) {
    const int kb = kt * 4 + 2 * half;
    if constexpr (Fout >= 16) {
      const int c = col0 + l16;
      b.x = Wt[(long long)kb * Fout + c];
      b.y = Wt[(long long)(kb + 1) * Fout + c];
    } else {  // Fout < 16: clamp column, mask value (no divergent loads)
      const int c = (l16 < Fout) ? l16 : 0;
      const float m = (l16 < Fout) ? 1.0f : 0.0f;
      b.x = Wt[(long long)kb * Fout + c] * m;
      b.y = Wt[(long long)(kb + 1) * Fout + c] * m;
    }
  } else {  // Fin == 1: only K=0 row exists (Fout >= 16 in this net)
    const float v = Wt[col0 + l16];
    b.x = (half == 0) ? v : 0.0f;
    b.y = 0.0f;
  }
  return b;
}

template <int Fin, int Fout, bool RELU>
__global__ void cheb_gemm_wmma_kernel(const float* __restrict__ T0,
                                      const float* __restrict__ T1,
                                      const float* __restrict__ T2,
                                      const float* __restrict__ W,  // [3, Fin, Fout]
                                      float* __restrict__ out, int N) {
  constexpr int KT = (Fin + 3) / 4;       // K-tiles of 4
  constexpr int TN = (Fout + 15) / 16;    // 16-wide column strips
  const int lane = threadIdx.x & 31;
  const int wave = threadIdx.x >> 5;
  const int half = lane >> 4;
  const int l16  = lane & 15;
  const int tilesM = N >> 4;              // N = 100000 = 16 * 6250
  const int jobsPerStrip = (tilesM + MT_PER_WAVE - 1) / MT_PER_WAVE;
  const int gw = blockIdx.x * WAVES_PER_BLOCK + wave;
  if (gw >= jobsPerStrip * TN) return;    // wave-uniform: EXEC all-ones at WMMA
  const int tn   = gw / jobsPerStrip;
  const int mjob = gw - tn * jobsPerStrip;
  const int col0 = tn * 16;

  // Hoist all B fragments (weights) into registers: reused across 4 M-tiles.
  v2f bfrag[3][KT];
#pragma unroll
  for (int t = 0; t < 3; ++t) {
    const float* __restrict__ Wt = W + (long long)t * Fin * Fout;
#pragma unroll
    for (int kt = 0; kt < KT; ++kt)
      bfrag[t][kt] = load_bfrag<Fin, Fout>(Wt, l16, half, kt, col0);
  }

  const float* Ts[3] = {T0, T1, T2};

#pragma unroll
  for (int i = 0; i < MT_PER_WAVE; ++i) {
    const int tm = mjob * MT_PER_WAVE + i;
    if (tm >= tilesM) break;              // wave-uniform
    const int row0 = tm << 4;
    const int arow = row0 + l16;
    v8f acc = {};
#pragma unroll
    for (int t = 0; t < 3; ++t) {
      const float* __restrict__ X = Ts[t];
#pragma unroll
      for (int kt = 0; kt < KT; ++kt) {
        const v2f a = load_afrag<Fin>(X, arow, kt, half);
        acc = __builtin_amdgcn_wmma_f32_16x16x4_f32(
            /*neg_a=*/false, a, /*neg_b=*/false, bfrag[t][kt],
            /*c_mod=*/(short)0, acc, /*reuse_a=*/false, /*reuse_b=*/false);
      }
    }
    const int m0 = row0 + 8 * half;
    if (Fout >= 16 || l16 < Fout) {
      const int n = (Fout >= 16) ? (col0 + l16) : l16;
#pragma unroll
      for (int r = 0; r < 8; ++r) {
        float v = acc[r];
        if (RELU) v = fmaxf(v, 0.0f);
        out[(long long)(m0 + r) * Fout + n] = v;
      }
    }
  }
}

// ---------- host side ----------
static inline int ilog2(int v) { int s = 0; while ((1 << s) < v) ++s; return s; }

extern "C" void kernel_launch(void* const* d_in, const int* in_sizes, int n_in,
                              void* d_out, int out_size, void* d_ws, size_t ws_size,
                              hipStream_t stream) {
  const float* x  = (const float*)d_in[0];
  const int*   ei = (const int*)d_in[1];
  const float* ea = (const float*)d_in[2];
  const float* W1 = (const float*)d_in[3];
  const float* W2 = (const float*)d_in[4];
  const float* W3 = (const float*)d_in[5];
  const float* W4 = (const float*)d_in[6];

  const int N = in_sizes[0];       // x is [N,1]
  const int E = in_sizes[2];       // edge_attr is [E]
  const int* src = ei;             // edge_index row 0
  const int* dst = ei + E;         // edge_index row 1

  // workspace carve-up (floats)
  float* ws   = (float*)d_ws;
  float* deg  = ws;  ws += N;                    // degree -> D^{-1/2} in place
  float* norm = ws;  ws += E;                    // per-edge coefficient
  float* bufA = ws;  ws += (long long)N * 64;    // ping
  float* bufB = ws;  ws += (long long)N * 64;    // Tx1 scratch
  float* bufC = ws;  ws += (long long)N * 64;    // Tx2 scratch
  float* bufH = ws;                              // pong

  auto zero = [&](float* p, long long n) {
    fill_zero_kernel<<<(int)((n + 255) / 256), 256, 0, stream>>>(p, n);
  };

  // --- graph normalization: norm = -D^{-1/2} A D^{-1/2} (lambda_max = 2) ---
  zero(deg, N);
  deg_kernel<<<(E + 255) / 256, 256, 0, stream>>>(src, ea, deg, E);
  dis_kernel<<<(N + 255) / 256, 256, 0, stream>>>(deg, N);
  norm_kernel<<<(E + 255) / 256, 256, 0, stream>>>(src, dst, ea, deg, norm, E);

  auto propagate = [&](const float* X, float* O, int F) {
    long long nf = (long long)N * F;
    zero(O, nf);
    if (F == 1) {
      propagate_f1_kernel<<<(E + 255) / 256, 256, 0, stream>>>(X, src, dst, norm, O, E);
    } else {
      const int LOG4 = ilog2(F) - 2;
      long long tot = (long long)E << LOG4;
      propagate_vec4_kernel<<<(int)((tot + 255) / 256), 256, 0, stream>>>(
          X, src, dst, norm, O, E, F, LOG4);
    }
  };

  auto gemm_grid = [&](int Fout) {
    const int tilesM = N >> 4;
    const int jobs = ((tilesM + MT_PER_WAVE - 1) / MT_PER_WAVE) * ((Fout + 15) / 16);
    return (jobs + WAVES_PER_BLOCK - 1) / WAVES_PER_BLOCK;
  };
  const dim3 gblk(32 * WAVES_PER_BLOCK);

  // --- layer 1: 1 -> 16, ReLU ---
  propagate(x, bufB, 1);
  propagate(bufB, bufC, 1);
  tx2_finish_kernel<<<(N + 255) / 256, 256, 0, stream>>>(bufC, x, N);
  cheb_gemm_wmma_kernel<1, 16, true><<<gemm_grid(16), gblk, 0, stream>>>(
      x, bufB, bufC, W1, bufA, N);

  // --- layer 2: 16 -> 32, ReLU ---
  propagate(bufA, bufB, 16);
  propagate(bufB, bufC, 16);
  tx2_finish_kernel<<<(int)(((long long)N * 16 + 255) / 256), 256, 0, stream>>>(
      bufC, bufA, (long long)N * 16);
  cheb_gemm_wmma_kernel<16, 32, true><<<gemm_grid(32), gblk, 0, stream>>>(
      bufA, bufB, bufC, W2, bufH, N);

  // --- layer 3: 32 -> 64, ReLU ---
  propagate(bufH, bufB, 32);
  propagate(bufB, bufC, 32);
  tx2_finish_kernel<<<(int)(((long long)N * 32 + 255) / 256), 256, 0, stream>>>(
      bufC, bufH, (long long)N * 32);
  cheb_gemm_wmma_kernel<32, 64, true><<<gemm_grid(64), gblk, 0, stream>>>(
      bufH, bufB, bufC, W3, bufA, N);

  // --- layer 4: 64 -> 2, no ReLU ---
  propagate(bufA, bufB, 64);
  propagate(bufB, bufC, 64);
  tx2_finish_kernel<<<(int)(((long long)N * 64 + 255) / 256), 256, 0, stream>>>(
      bufC, bufA, (long long)N * 64);
  cheb_gemm_wmma_kernel<64, 2, false><<<gemm_grid(2), gblk, 0, stream>>>(
      bufA, bufB, bufC, W4, (float*)d_out, N);
}